// PhyloDisentangler_73882027426325
// MI455X (gfx1250) — compile-verified
//
#include <hip/hip_runtime.h>
#include <hip/hip_bf16.h>

typedef __attribute__((ext_vector_type(2))) float v2f;
typedef __attribute__((ext_vector_type(8))) float v8f;

__device__ __forceinline__ float silu_f(float x) {
    // x * sigmoid(x); v_rcp_f32 instead of precise division (epilogue only)
    return x * __builtin_amdgcn_rcpf(1.0f + __expf(-x));
}

__device__ __forceinline__ v8f wmma4(v2f a, v2f b, v8f c) {
    // D(16x16 f32) = A(16x4 f32) * B(4x16 f32) + C
    return __builtin_amdgcn_wmma_f32_16x16x4_f32(
        /*neg_a=*/false, a, /*neg_b=*/false, b,
        /*c_mod=*/(short)0, c, /*reuse_a=*/false, /*reuse_b=*/false);
}

// ---------------------------------------------------------------------------
// K1: conv_in  h[b][o][p] = silu( sum_c W[o][c] * X[b][c][p] + bias[o] )
// One wave per 16x16 (o,p) tile, K=256 in steps of 4 via f32 WMMA.
// ---------------------------------------------------------------------------
__global__ __launch_bounds__(256) void k_conv_in(
    const float* __restrict__ W,    // (256,256) [o][c]
    const float* __restrict__ bias, // (256)
    const float* __restrict__ X,    // (32,256,256) [b][c][p]
    float* __restrict__ Y)          // (32,256,256) [b][o][p]
{
    int wv   = (blockIdx.x * 256 + threadIdx.x) >> 5;   // 0..8191
    int b    = wv >> 8;
    int rem  = wv & 255;
    int ot   = rem >> 4;
    int pt   = rem & 15;
    int lane = threadIdx.x & 31;
    int half = lane >> 4;
    int lh   = lane & 15;

    const float* Wrow = W + (ot * 16 + lh) * 256;
    const float* Xb   = X + (size_t)b * 65536;
    int pcol = pt * 16 + lh;

    v8f acc = {};
    for (int k0 = 0; k0 < 256; k0 += 4) {
        int ka = k0 + 2 * half;
        v2f a;  a.x  = Wrow[ka];               a.y  = Wrow[ka + 1];
        v2f bb; bb.x = Xb[ka * 256 + pcol];    bb.y = Xb[(ka + 1) * 256 + pcol];
        acc = wmma4(a, bb, acc);
    }
#pragma unroll
    for (int r = 0; r < 8; r++) {
        int m = ot * 16 + r + 8 * half;
        Y[(size_t)b * 65536 + m * 256 + pcol] = silu_f(acc[r] + bias[m]);
    }
}

// ---------------------------------------------------------------------------
// K2: per-batch mean / rstd over h[:, :128, :, :]  (32768 elems per batch)
// ---------------------------------------------------------------------------
__global__ __launch_bounds__(256) void k_ln_stats(
    const float* __restrict__ h, float* __restrict__ mu, float* __restrict__ rstd)
{
    __shared__ float ss[256], ss2[256];
    int b = blockIdx.x;
    const float* p = h + (size_t)b * 65536;
    float s = 0.f, s2 = 0.f;
    for (int i = threadIdx.x; i < 32768; i += 256) {
        float x = p[i]; s += x; s2 += x * x;
    }
    ss[threadIdx.x] = s; ss2[threadIdx.x] = s2;
    __syncthreads();
    for (int o = 128; o > 0; o >>= 1) {
        if (threadIdx.x < o) {
            ss[threadIdx.x]  += ss[threadIdx.x + o];
            ss2[threadIdx.x] += ss2[threadIdx.x + o];
        }
        __syncthreads();
    }
    if (threadIdx.x == 0) {
        float m   = ss[0] * (1.0f / 32768.0f);
        float var = ss2[0] * (1.0f / 32768.0f) - m * m;
        mu[b]   = m;
        rstd[b] = rsqrtf(var + 1e-5f);
    }
}

// ---------------------------------------------------------------------------
// K3: zn[b][k] = (h[b][k] - mu[b]) * rstd[b] * g[k] + be[k]   (k = c*256+p, c<128)
// ---------------------------------------------------------------------------
__global__ __launch_bounds__(256) void k_ln_apply(
    const float* __restrict__ h, const float* __restrict__ mu,
    const float* __restrict__ rstd, const float* __restrict__ g,
    const float* __restrict__ be, float* __restrict__ zn)
{
    int i = blockIdx.x * 256 + threadIdx.x;   // over 32*32768
    int b = i >> 15;
    int k = i & 32767;
    zn[i] = (h[(size_t)b * 65536 + k] - mu[b]) * rstd[b] * g[k] + be[k];
}

// ---------------------------------------------------------------------------
// Split-K GEMM partial:  P[s] = A(32 x klen) @ W(klen x N)  for K-chunk s.
// grid.x = 2*(N/128) tiles, grid.y = S splits. 8 waves/block share one
// 16-row A chunk in LDS; each wave owns a private 16-col stripe of W so every
// weight element is read by exactly two blocks (mt = 0 and mt = 1) across the grid.
// ---------------------------------------------------------------------------
__global__ __launch_bounds__(256) void k_gemm32_part(
    const float* __restrict__ A, const float* __restrict__ W,
    float* __restrict__ P, int K, int N, int klen)
{
    __shared__ float As[16 * 256];
    int mt  = blockIdx.x & 1;
    int ng  = blockIdx.x >> 1;
    int sp  = blockIdx.y;
    int kbase = sp * klen;
    int tid = threadIdx.x;
    int lane = tid & 31, wvid = tid >> 5;
    int half = lane >> 4, lh = lane & 15;
    int ncol = ng * 128 + wvid * 16 + lh;

    v8f acc = {};
    for (int kc = kbase; kc < kbase + klen; kc += 256) {
#pragma unroll
        for (int t = 0; t < 16; t++)
            As[t * 256 + tid] = A[(size_t)(mt * 16 + t) * K + kc + tid];
        __syncthreads();

        const float* as = As + lh * 256;
        for (int ks = 0; ks < 256; ks += 4) {
            int kk = ks + 2 * half;
            v2f a;  a.x = as[kk]; a.y = as[kk + 1];
            size_t kg = (size_t)(kc + kk);
            v2f bb; bb.x = W[kg * N + ncol]; bb.y = W[(kg + 1) * N + ncol];
            acc = wmma4(a, bb, acc);
        }
        __syncthreads();
    }
#pragma unroll
    for (int r = 0; r < 8; r++) {
        int m = mt * 16 + r + 8 * half;
        P[(size_t)sp * 32 * N + (size_t)m * N + ncol] = acc[r];
    }
}

// ---------------------------------------------------------------------------
// Split-K reduce: Y[m][n] = silu( sum_s P[s][m][n] + bias[n] ).
// Fixed summation order -> bit-deterministic across replays.
// ---------------------------------------------------------------------------
__global__ __launch_bounds__(256) void k_gemm32_reduce(
    const float* __restrict__ P, const float* __restrict__ bias,
    float* __restrict__ Y, int N, int S)
{
    int i = blockIdx.x * 256 + threadIdx.x;   // over 32*N
    int n = i & (N - 1);                      // N is a power of two
    float s = 0.f;
    for (int j = 0; j < S; j++) s += P[(size_t)j * 32 * N + i];
    Y[i] = silu_f(s + bias[n]);
}

// ---------------------------------------------------------------------------
// K5: codebook squared norms
// ---------------------------------------------------------------------------
__global__ __launch_bounds__(256) void k_cb_norm(
    const float* __restrict__ cb, float* __restrict__ cbn)
{
    int n = blockIdx.x * 256 + threadIdx.x;   // 1024 total
    const float* r = cb + (size_t)n * 128;
    float s = 0.f;
    for (int e = 0; e < 128; e++) { float x = r[e]; s += x * x; }
    cbn[n] = s;
}

// ---------------------------------------------------------------------------
// K6: G(1024x1024) = zf(1024x128) @ cb^T  via WMMA.
// zf[b*32+s][e] == z[b][e*32+s] so A tiles load straight from z (no reshape).
// B tile = one contiguous 8-byte load per lane from the codebook row.
// ---------------------------------------------------------------------------
__global__ __launch_bounds__(256) void k_vq_gemm(
    const float* __restrict__ z, const float* __restrict__ cb,
    float* __restrict__ G)
{
    int wv = (blockIdx.x * 256 + threadIdx.x) >> 5;  // 0..4095
    int rt = wv >> 6, nt = wv & 63;
    int lane = threadIdx.x & 31, half = lane >> 4, lh = lane & 15;
    int b = rt >> 1;
    int s0 = (rt & 1) * 16;
    const float* zb  = z + (size_t)b * 4096;
    const float* cbr = cb + (size_t)(nt * 16 + lh) * 128;

    v8f acc = {};
    for (int k0 = 0; k0 < 128; k0 += 4) {
        int kk = k0 + 2 * half;
        v2f a;  a.x = zb[kk * 32 + s0 + lh];  a.y = zb[(kk + 1) * 32 + s0 + lh];
        v2f bb; bb.x = cbr[kk];               bb.y = cbr[kk + 1];
        acc = wmma4(a, bb, acc);
    }
#pragma unroll
    for (int r = 0; r < 8; r++) {
        int row = rt * 16 + r + 8 * half;
        G[(size_t)row * 1024 + nt * 16 + lh] = acc[r];
    }
}

// ---------------------------------------------------------------------------
// K7: per-row argmin (score = ||cb||^2 - 2*G), gather codebook row back into
// z-layout, and per-row squared error for the VQ loss. First-index tie-break
// to match jnp.argmin.
// ---------------------------------------------------------------------------
__global__ __launch_bounds__(128) void k_vq_select(
    const float* __restrict__ G, const float* __restrict__ cbn,
    const float* __restrict__ z, const float* __restrict__ cb,
    float* __restrict__ zq, float* __restrict__ rowloss)
{
    __shared__ float sv[128];
    __shared__ int   si[128];
    __shared__ int   bestIdx;
    int r = blockIdx.x, t = threadIdx.x;
    const float* Gr = G + (size_t)r * 1024;

    float best = 3.4e38f; int bi = 0;
    for (int n = t; n < 1024; n += 128) {
        float sc = cbn[n] - 2.0f * Gr[n];
        if (sc < best) { best = sc; bi = n; }
    }
    sv[t] = best; si[t] = bi;
    __syncthreads();
    for (int o = 64; o > 0; o >>= 1) {
        if (t < o) {
            float a = sv[t], b2 = sv[t + o];
            if (b2 < a || (b2 == a && si[t + o] < si[t])) { sv[t] = b2; si[t] = si[t + o]; }
        }
        __syncthreads();
    }
    if (t == 0) bestIdx = si[0];
    __syncthreads();
    int idx = bestIdx;

    int b = r >> 5, s = r & 31;
    float q  = cb[(size_t)idx * 128 + t];
    float zp = z[(size_t)b * 4096 + t * 32 + s];
    zq[(size_t)b * 4096 + t * 32 + s] = q;
    float d = q - zp;
    sv[t] = d * d;
    __syncthreads();
    for (int o = 64; o > 0; o >>= 1) {
        if (t < o) sv[t] += sv[t + o];
        __syncthreads();
    }
    if (t == 0) rowloss[r] = sv[0];
}

// ---------------------------------------------------------------------------
// K8: final deterministic loss reduce:  1.25 * mean over 131072 elements
// ---------------------------------------------------------------------------
__global__ __launch_bounds__(256) void k_vq_loss(
    const float* __restrict__ rowloss, float* __restrict__ out_loss)
{
    __shared__ float sv[256];
    float s = 0.f;
    for (int i = threadIdx.x; i < 1024; i += 256) s += rowloss[i];
    sv[threadIdx.x] = s;
    __syncthreads();
    for (int o = 128; o > 0; o >>= 1) {
        if (threadIdx.x < o) sv[threadIdx.x] += sv[threadIdx.x + o];
        __syncthreads();
    }
    if (threadIdx.x == 0) *out_loss = 1.25f * sv[0] * (1.0f / 131072.0f);
}

// ---------------------------------------------------------------------------
// K10: conv_out on concat([zq2, h_img]):  c<128 -> g2, c>=128 -> h.
// k0 is 4-aligned so each WMMA reads entirely from one source.
// ---------------------------------------------------------------------------
__global__ __launch_bounds__(256) void k_conv_out(
    const float* __restrict__ W,    // (256,256) [o][c]
    const float* __restrict__ bias, // (256)
    const float* __restrict__ g2,   // (32,32768)  [b][c*256+p], c<128
    const float* __restrict__ h,    // (32,256,256)
    float* __restrict__ out)        // (32,256,256)
{
    int wv   = (blockIdx.x * 256 + threadIdx.x) >> 5;
    int b    = wv >> 8;
    int rem  = wv & 255;
    int ot   = rem >> 4;
    int pt   = rem & 15;
    int lane = threadIdx.x & 31;
    int half = lane >> 4;
    int lh   = lane & 15;

    const float* Wrow = W + (ot * 16 + lh) * 256;
    const float* lo = g2 + (size_t)b * 32768;
    const float* hi = h  + (size_t)b * 65536;
    int pcol = pt * 16 + lh;

    v8f acc = {};
    for (int k0 = 0; k0 < 256; k0 += 4) {
        const float* src = (k0 < 128) ? lo : hi;
        int ka = k0 + 2 * half;
        v2f a;  a.x  = Wrow[ka];                a.y  = Wrow[ka + 1];
        v2f bb; bb.x = src[ka * 256 + pcol];    bb.y = src[(ka + 1) * 256 + pcol];
        acc = wmma4(a, bb, acc);
    }
#pragma unroll
    for (int r = 0; r < 8; r++) {
        int m = ot * 16 + r + 8 * half;
        out[(size_t)b * 65536 + m * 256 + pcol] = silu_f(acc[r] + bias[m]);
    }
}

// ---------------------------------------------------------------------------
extern "C" void kernel_launch(void* const* d_in, const int* in_sizes, int n_in,
                              void* d_out, int out_size, void* d_ws, size_t ws_size,
                              hipStream_t stream) {
    const float* input      = (const float*)d_in[0];
    const float* conv_in_w  = (const float*)d_in[1];
    const float* conv_in_b  = (const float*)d_in[2];
    const float* ln_g       = (const float*)d_in[3];
    const float* ln_b       = (const float*)d_in[4];
    const float* w1         = (const float*)d_in[5];
    const float* b1         = (const float*)d_in[6];
    const float* codebook   = (const float*)d_in[7];
    const float* w2         = (const float*)d_in[8];
    const float* b2         = (const float*)d_in[9];
    const float* conv_out_w = (const float*)d_in[10];
    const float* conv_out_b = (const float*)d_in[11];
    float* out = (float*)d_out;

    // workspace layout (floats)
    float* h    = (float*)d_ws;        // 32*256*256 = 2,097,152
    float* zn   = h    + 2097152;      // 32*32768   = 1,048,576
    float* z    = zn   + 1048576;      // 32*4096    = 131,072
    float* zq   = z    + 131072;       // 131,072
    float* G    = zq   + 131072;       // 1024*1024  = 1,048,576
    float* g2   = G    + 1048576;      // 1,048,576
    float* P    = g2   + 1048576;      // split-K partials: max(8*32*4096, 4*32*32768) = 4,194,304
    float* mu   = P    + 4194304;      // 32
    float* rstd = mu   + 32;           // 32
    float* cbn  = rstd + 32;           // 1024
    float* rl   = cbn  + 1024;         // 1024

    k_conv_in   <<<1024, 256, 0, stream>>>(conv_in_w, conv_in_b, input, h);
    k_ln_stats  <<<32,   256, 0, stream>>>(h, mu, rstd);
    k_ln_apply  <<<4096, 256, 0, stream>>>(h, mu, rstd, ln_g, ln_b, zn);

    // GEMM1: zn(32x32768) @ w1(32768x4096), split K into 8 -> 512 blocks
    k_gemm32_part  <<<dim3(64, 8), 256, 0, stream>>>(zn, w1, P, 32768, 4096, 4096);
    k_gemm32_reduce<<<512, 256, 0, stream>>>(P, b1, z, 4096, 8);

    k_cb_norm   <<<4,    256, 0, stream>>>(codebook, cbn);
    k_vq_gemm   <<<512,  256, 0, stream>>>(z, codebook, G);
    k_vq_select <<<1024, 128, 0, stream>>>(G, cbn, z, codebook, zq, rl);
    k_vq_loss   <<<1,    256, 0, stream>>>(rl, out + 2097152);

    // GEMM2: zq(32x4096) @ w2(4096x32768), split K into 4 -> 2048 blocks
    k_gemm32_part  <<<dim3(512, 4), 256, 0, stream>>>(zq, w2, P, 4096, 32768, 1024);
    k_gemm32_reduce<<<4096, 256, 0, stream>>>(P, b2, g2, 32768, 4);

    k_conv_out  <<<1024, 256, 0, stream>>>(conv_out_w, conv_out_b, g2, h, out);
}